// AttentionBlockCached_33268816674850
// MI455X (gfx1250) — compile-verified
//
#include <hip/hip_runtime.h>
#include <hip/hip_bf16.h>
#include <math.h>

// ---- problem constants ----
#define B_    8
#define T_    16
#define HID_  4096
#define HQ_   64
#define HKV_  8
#define D_    64
#define TC_   4096
#define TTOT_ (TC_ + T_)          // 4112
#define WIN_  128
#define QKV_DIM_ 5120             // D*(HQ+2*HKV)
#define SM_SCALE_ 0.125f          // 1/sqrt(64)

typedef float v2f __attribute__((ext_vector_type(2)));
typedef float v8f __attribute__((ext_vector_type(8)));

// -------------------------------------------------------------------
// 1) RMSNorm: h[row] = x[row] * rsqrt(mean(x^2)+eps) * w
// -------------------------------------------------------------------
__global__ void rmsnorm_kernel(const float* __restrict__ x,
                               const float* __restrict__ w,
                               float* __restrict__ h) {
  const int row = blockIdx.x;               // 0..127
  const int tid = threadIdx.x;              // 256 threads
  const float* xr = x + (size_t)row * HID_;
  float ss = 0.f;
  for (int i = tid; i < HID_; i += 256) { float v = xr[i]; ss += v * v; }
  __shared__ float red[256];
  red[tid] = ss;
  __syncthreads();
  for (int s = 128; s > 0; s >>= 1) {
    if (tid < s) red[tid] += red[tid + s];
    __syncthreads();
  }
  const float scale = rsqrtf(red[0] / (float)HID_ + 1e-5f);
  float* hr = h + (size_t)row * HID_;
  for (int i = tid; i < HID_; i += 256) hr[i] = xr[i] * scale * w[i];
}

// -------------------------------------------------------------------
// 2) fp32 WMMA GEMM: out[M=128,N] = A[128,4096] @ W[N,4096]^T + bias (+resid)
//    One wave per (16-wide N strip, 64-row M group): 4 m-tiles per wave.
//    Software-pipelined: next k-step's A/B fragments are loaded into a second
//    register buffer while the current 8 WMMAs run, so global latency
//    overlaps the matrix pipe instead of serializing in front of it.
//    W fits in L2 (84/67 MB < 192 MB), so the 2x W re-read from the two
//    M groups costs L2 bandwidth only; HBM still streams W once.
// -------------------------------------------------------------------
#define MT_ 4                              // m-tiles per wave
#define A_MT_STRIDE_ (16 * 4096)           // floats between m-tiles in A

__global__ void wmma_gemm_kernel(const float* __restrict__ A,
                                 const float* __restrict__ W,
                                 const float* __restrict__ bias,
                                 const float* __restrict__ resid,
                                 float* __restrict__ out, int N) {
  const int lane   = threadIdx.x & 31;
  const int wave   = threadIdx.x >> 5;      // 4 waves per block
  const int ntile  = blockIdx.x * 4 + wave;
  const int n0     = ntile * 16;
  const int mgroup = blockIdx.y;            // 0 or 1 (rows 0..63 / 64..127)
  const int half   = lane >> 4;             // 0: K pair {0,1}, 1: {2,3}
  const int l16    = lane & 15;

  const float* Ap = A + (size_t)(mgroup * 64 + l16) * 4096 + half * 2;
  const float* Wp = W + (size_t)(n0 + l16) * 4096 + half * 2;

  v8f acc[MT_] = {};

  // ---- pipeline prologue: load k=0 fragments ----
  v2f cb0 = *(const v2f*)(Wp);
  v2f cb1 = *(const v2f*)(Wp + 4);
  v2f ca0[MT_], ca1[MT_];
#pragma unroll
  for (int mt = 0; mt < MT_; ++mt) {
    ca0[mt] = *(const v2f*)(Ap + mt * A_MT_STRIDE_);
    ca1[mt] = *(const v2f*)(Ap + mt * A_MT_STRIDE_ + 4);
  }

#pragma unroll 2
  for (int k = 0; k < 4096; k += 8) {
    const int kn = (k + 8) & 4095;          // wraps to 0 on last iter (no OOB)
    // ---- issue next-step loads (one clause, overlapped with WMMAs) ----
    const v2f nb0 = *(const v2f*)(Wp + kn);
    const v2f nb1 = *(const v2f*)(Wp + kn + 4);
    v2f na0[MT_], na1[MT_];
#pragma unroll
    for (int mt = 0; mt < MT_; ++mt) {
      na0[mt] = *(const v2f*)(Ap + mt * A_MT_STRIDE_ + kn);
      na1[mt] = *(const v2f*)(Ap + mt * A_MT_STRIDE_ + kn + 4);
    }
    // stream W ahead of the pipelined loads (global_prefetch_b8)
    __builtin_prefetch(Wp + ((k + 512) & 4095), 0, 1);

    // ---- compute on current buffers ----
#pragma unroll
    for (int mt = 0; mt < MT_; ++mt) {
      acc[mt] = __builtin_amdgcn_wmma_f32_16x16x4_f32(
          false, ca0[mt], false, cb0, (short)0, acc[mt], false, false);
      acc[mt] = __builtin_amdgcn_wmma_f32_16x16x4_f32(
          false, ca1[mt], false, cb1, (short)0, acc[mt], false, false);
    }
    // ---- rotate buffers (dissolves under unroll 2) ----
    cb0 = nb0; cb1 = nb1;
#pragma unroll
    for (int mt = 0; mt < MT_; ++mt) { ca0[mt] = na0[mt]; ca1[mt] = na1[mt]; }
  }

  const int n = n0 + l16;
  const float bn = bias[n];
#pragma unroll
  for (int mt = 0; mt < MT_; ++mt) {
#pragma unroll
    for (int v = 0; v < 8; ++v) {
      const int m = mgroup * 64 + mt * 16 + half * 8 + v;
      float r = acc[mt][v] + bn;
      if (resid) r += resid[(size_t)m * N + n];
      out[(size_t)m * N + n] = r;
    }
  }
}

// -------------------------------------------------------------------
// 3) cache concat: copy cache_k/cache_v into K/V output (b-stride TC -> TC+T)
// -------------------------------------------------------------------
__global__ void copy_cache_kernel(const float4* __restrict__ ck,
                                  const float4* __restrict__ cv,
                                  float4* __restrict__ Kout,
                                  float4* __restrict__ Vout) {
  const size_t per_b4 = (size_t)TC_ * HKV_ * D_ / 4;     // 524288
  const size_t dst_b4 = (size_t)TTOT_ * HKV_ * D_ / 4;   // 526336
  const size_t total  = (size_t)B_ * per_b4;
  for (size_t i = (size_t)blockIdx.x * blockDim.x + threadIdx.x; i < total;
       i += (size_t)gridDim.x * blockDim.x) {
    const size_t b = i / per_b4, rem = i - b * per_b4;
    const size_t d = b * dst_b4 + rem;
    Kout[d] = ck[i];
    Vout[d] = cv[i];
  }
}

// -------------------------------------------------------------------
// 4) RoPE on q (in place in qkv) and k (written to K slot TC+t); v appended.
// -------------------------------------------------------------------
__global__ void rope_append_kernel(float* __restrict__ qkv,
                                   float* __restrict__ Kout,
                                   float* __restrict__ Vout) {
  const int row = blockIdx.x;               // b*T + t
  const int b = row / T_, t = row % T_;
  const int tid = threadIdx.x;              // 256
  const float pos = (float)(TC_ + t);
  float* base = qkv + (size_t)row * QKV_DIM_;

  // q: 64 heads x 32 rotation pairs = 2048 units
#pragma unroll
  for (int it = 0; it < 8; ++it) {
    const int u = tid + it * 256;
    const int hq = u >> 5, i = u & 31;
    const float invf = powf(150000.0f, -(float)i / 32.0f);
    const float ang = pos * invf;
    const float c = cosf(ang), s = sinf(ang);
    float* p = base + hq * D_;
    const float x1 = p[i], x2 = p[i + 32];
    p[i]      = x1 * c - x2 * s;
    p[i + 32] = x2 * c + x1 * s;
  }
  // k: 8 heads x 32 pairs = 256 units
  {
    const int hkv = tid >> 5, i = tid & 31;
    const float invf = powf(150000.0f, -(float)i / 32.0f);
    const float ang = pos * invf;
    const float c = cosf(ang), s = sinf(ang);
    const float* p = base + HQ_ * D_ + hkv * D_;
    const float x1 = p[i], x2 = p[i + 32];
    float* kd = Kout + (((size_t)b * TTOT_ + TC_ + t) * HKV_ + hkv) * D_;
    kd[i]      = x1 * c - x2 * s;
    kd[i + 32] = x2 * c + x1 * s;
  }
  // v: 512 elements
#pragma unroll
  for (int it = 0; it < 2; ++it) {
    const int u = tid + it * 256;
    const int hkv = u >> 6, d = u & 63;
    Vout[(((size_t)b * TTOT_ + TC_ + t) * HKV_ + hkv) * D_ + d] =
        base[(HQ_ + HKV_) * D_ + u];
  }
}

// -------------------------------------------------------------------
// 5) sliding-window attention with sink logit. 1 wave per (b,hq,t).
// -------------------------------------------------------------------
__global__ void attn_kernel(const float* __restrict__ qkv,
                            const float* __restrict__ Kout,
                            const float* __restrict__ Vout,
                            const float* __restrict__ sinks,
                            float* __restrict__ attn) {
  const int lane = threadIdx.x;             // 32 threads
  const int idx = blockIdx.x;               // ((b*HQ + hq)*T + t)
  const int t = idx % T_;
  const int bh = idx / T_;
  const int hq = bh % HQ_;
  const int b = bh / HQ_;
  const int hkv = hq >> 3;                  // hq / Q_MULT
  const int row = b * T_ + t;
  const int pos_q = TC_ + t;
  const int kstart = pos_q - WIN_;          // >= 0 (TC >= WIN)
  const int count = WIN_ + 1;               // 129 keys

  __shared__ float qs[D_];
  const float* qp = qkv + (size_t)row * QKV_DIM_ + hq * D_;
  qs[lane] = qp[lane];
  qs[lane + 32] = qp[lane + 32];
  __syncthreads();

  const float sink = sinks[hq];
  float sreg[5];
  float mx = sink;
#pragma unroll
  for (int r = 0; r < 5; ++r) {
    const int j = r * 32 + lane;
    float s = -3.4e38f;
    if (j < count) {
      const float* kp = Kout + (((size_t)b * TTOT_ + kstart + j) * HKV_ + hkv) * D_;
      float acc = 0.f;
#pragma unroll
      for (int d = 0; d < D_; ++d) acc += qs[d] * kp[d];
      s = acc * SM_SCALE_;
    }
    sreg[r] = s;
    mx = fmaxf(mx, s);
  }
#pragma unroll
  for (int m = 16; m >= 1; m >>= 1) mx = fmaxf(mx, __shfl_xor(mx, m, 32));

  float sum = 0.f;
#pragma unroll
  for (int r = 0; r < 5; ++r) {
    const float p = (sreg[r] > -1e37f) ? __expf(sreg[r] - mx) : 0.f;
    sreg[r] = p;
    sum += p;
  }
#pragma unroll
  for (int m = 16; m >= 1; m >>= 1) sum += __shfl_xor(sum, m, 32);
  const float inv = 1.0f / (sum + __expf(sink - mx));

  float a0 = 0.f, a1 = 0.f;
#pragma unroll 1
  for (int r = 0; r < 4; ++r) {
    for (int jj = 0; jj < 32; ++jj) {
      const float w = __shfl(sreg[r], jj, 32) * inv;
      const int pos = kstart + r * 32 + jj;
      const float* vp = Vout + (((size_t)b * TTOT_ + pos) * HKV_ + hkv) * D_;
      a0 += w * vp[lane];
      a1 += w * vp[lane + 32];
    }
  }
  { // last key j = 128 -> pos == pos_q
    const float w = __shfl(sreg[4], 0, 32) * inv;
    const float* vp = Vout + (((size_t)b * TTOT_ + pos_q) * HKV_ + hkv) * D_;
    a0 += w * vp[lane];
    a1 += w * vp[lane + 32];
  }

  attn[(size_t)row * (HQ_ * D_) + hq * D_ + lane] = a0;
  attn[(size_t)row * (HQ_ * D_) + hq * D_ + lane + 32] = a1;
}

// -------------------------------------------------------------------
extern "C" void kernel_launch(void* const* d_in, const int* in_sizes, int n_in,
                              void* d_out, int out_size, void* d_ws, size_t ws_size,
                              hipStream_t stream) {
  const float* x       = (const float*)d_in[0];
  const float* cache_k = (const float*)d_in[1];
  const float* cache_v = (const float*)d_in[2];
  // d_in[3] = cache_position (int scalar, == TC_)
  const float* sinks   = (const float*)d_in[4];
  const float* norm_w  = (const float*)d_in[5];
  const float* qkv_w   = (const float*)d_in[6];
  const float* qkv_b   = (const float*)d_in[7];
  const float* out_w   = (const float*)d_in[8];
  const float* out_b   = (const float*)d_in[9];

  float* out  = (float*)d_out;                              // (B,T,HID)
  float* Kout = out + (size_t)B_ * T_ * HID_;               // (B,TTOT,HKV,D)
  float* Vout = Kout + (size_t)B_ * TTOT_ * HKV_ * D_;      // (B,TTOT,HKV,D)

  float* h    = (float*)d_ws;                               // 128*4096
  float* qkv  = h + (size_t)B_ * T_ * HID_;                 // 128*5120
  float* attn = h;  // reuse: h dead after QKV GEMM

  rmsnorm_kernel<<<B_ * T_, 256, 0, stream>>>(x, norm_w, h);

  // QKV: N = 5120 -> 320 n-tiles -> 80 blocks x 2 m-groups, 4 waves/block
  wmma_gemm_kernel<<<dim3(QKV_DIM_ / 64, 2), 128, 0, stream>>>(
      h, qkv_w, qkv_b, nullptr, qkv, QKV_DIM_);

  copy_cache_kernel<<<2048, 256, 0, stream>>>(
      (const float4*)cache_k, (const float4*)cache_v,
      (float4*)Kout, (float4*)Vout);

  rope_append_kernel<<<B_ * T_, 256, 0, stream>>>(qkv, Kout, Vout);

  attn_kernel<<<B_ * HQ_ * T_, 32, 0, stream>>>(qkv, Kout, Vout, sinks, attn);

  // Out projection: N = 4096 -> 256 n-tiles -> 64 blocks x 2 m-groups
  wmma_gemm_kernel<<<dim3(HID_ / 64, 2), 128, 0, stream>>>(
      attn, out_w, out_b, x, out, HID_);
}